// RetinaLayer_64415919505700
// MI455X (gfx1250) — compile-verified
//
#include <hip/hip_runtime.h>
#include <stdint.h>

#define NB   4
#define NAT  9
#define NH   64
#define NW   64
#define AA   (NAT*NH*NW)      /* 36864 anchors per image */
#define NGT  32
#define NCLS 80

#define POS_THR 0.5f
#define NEG_THR 0.4f
#define HIGH_ENOUGH  2.9444389791664403f   /* log(0.95/0.05) */
#define LOW_ENOUGH  -4.59511985013459f     /* log(0.01/0.99) */

#define K1_BLOCKS (NB * (AA/256))   /* 576  */
#define K2_BLOCKS 2048

/* anchor (w,h) per anchor-type: BASE=32, scales {1, 2^(1/3), 2^(2/3)},
   ratios {(1,1),(1.4,0.7),(0.7,1.4)} — double-precision products rounded to f32,
   matching jnp.asarray(..., float32). */
__constant__ float c_aw[9] = {
    32.0f, 44.8f, 22.4f,
    40.31747359663594f, 56.44446303529032f, 28.22223151764516f,
    50.79683366298238f, 71.11556712817533f, 35.557783564087666f };
__constant__ float c_ah[9] = {
    32.0f, 22.4f, 44.8f,
    40.31747359663594f, 28.22223151764516f, 56.44446303529032f,
    50.79683366298238f, 35.557783564087666f, 71.11556712817533f };

/* ---- CDNA5 async global->LDS copy (16B per active lane) + split-counter wait ---- */
__device__ __forceinline__ void async_copy_b128(void* lds_dst, const void* gsrc) {
    uint32_t ldsoff = (uint32_t)(uintptr_t)lds_dst;   /* low 32 bits of flat LDS ptr = LDS offset */
    uint64_t gaddr  = (uint64_t)(uintptr_t)gsrc;
    asm volatile("global_load_async_to_lds_b128 %0, %1, off"
                 :: "v"(ldsoff), "v"(gaddr) : "memory");
}
__device__ __forceinline__ void wait_async0() {
    asm volatile("s_wait_asynccnt 0" ::: "memory");
}

__device__ __forceinline__ float smooth_l1(float d) {  /* beta = 0.1 */
    float ad = fabsf(d);
    return (ad < 0.1f) ? ((0.5f * d * d) / 0.1f) : (ad - 0.05f);
}

__device__ __forceinline__ float block_reduce_256(float v, float* sh) {
    int t = threadIdx.x;
    sh[t] = v; __syncthreads();
    #pragma unroll
    for (int s = 128; s > 0; s >>= 1) {
        if (t < s) sh[t] += sh[t + s];
        __syncthreads();
    }
    return sh[0];
}

/* Kernel 1: per-anchor matching + regression loss + meta word */
__global__ __launch_bounds__(256)
void k_anchor(const float4* __restrict__ t_xywh,     /* [NB*AA] float4 */
              const float*  __restrict__ gt_bboxes,  /* [NB*NGT*4]     */
              const int*    __restrict__ gt_cats,    /* [NB*NGT]       */
              uint16_t*     __restrict__ meta,       /* [NB*AA]        */
              float*        __restrict__ partial)    /* [K1_BLOCKS]    */
{
    __shared__ __align__(16) float sh_gt[NGT * 4];   /* 512 B: one batch's GT boxes */
    __shared__ float sh_red[256];

    const int blocksPerBatch = AA / 256;             /* 144 */
    const int b = blockIdx.x / blocksPerBatch;
    const int a = (blockIdx.x % blocksPerBatch) * 256 + threadIdx.x;

    /* wave 0 stages the 512-byte GT table via CDNA5 async global->LDS DMA */
    if (threadIdx.x < 32) {
        const char* src = (const char*)gt_bboxes + (size_t)b * (NGT * 4 * 4) + threadIdx.x * 16;
        async_copy_b128((char*)sh_gt + threadIdx.x * 16, src);
        wait_async0();
    }
    __syncthreads();

    /* anchor geometry from index: a = type*4096 + y*64 + x */
    const int   x  = a & (NW - 1);
    const int   y  = (a >> 6) & (NH - 1);
    const int   at = a >> 12;
    const float acx = (x + 0.5f) * 8.0f;
    const float acy = (y + 0.5f) * 8.0f;
    const float aw = c_aw[at], ah = c_ah[at];
    const float alx = acx - 0.5f * aw, aly = acy - 0.5f * ah;
    const float arx = acx + 0.5f * aw, ary = acy + 0.5f * ah;
    const float areaA = aw * ah;

    const float4* g4 = (const float4*)sh_gt;
    float best = -1.0f; int bi = 0;
    #pragma unroll 8
    for (int g = 0; g < NGT; ++g) {
        float4 gb = g4[g];                            /* LDS broadcast */
        float blx = gb.x - 0.5f * gb.z, bly = gb.y - 0.5f * gb.w;
        float brx = gb.x + 0.5f * gb.z, bry = gb.y + 0.5f * gb.w;
        float lx = fmaxf(alx, blx), ly = fmaxf(aly, bly);
        float rx = fminf(arx, brx), ry = fminf(ary, bry);
        float iw = fmaxf(rx - lx, 0.0f), ih = fmaxf(ry - ly, 0.0f);
        float inter = iw * ih;
        float iou = inter / (areaA + gb.z * gb.w - inter);
        if (iou > best) { best = iou; bi = g; }       /* strict > keeps first (argmax) */
    }

    const bool pos = best > POS_THR;
    const bool neg = best < NEG_THR;
    const int  cat = gt_cats[b * NGT + bi];
    meta[(size_t)b * AA + a] =
        (uint16_t)((cat & 0xFF) | (pos ? 256u : 0u) | (neg ? 512u : 0u));

    float acc = 0.0f;
    if (pos) {
        float4 gb = g4[bi];
        float4 tv = t_xywh[(size_t)b * AA + a];       /* coalesced b128 */
        float tx = (gb.x - acx) / aw;
        float ty = (gb.y - acy) / ah;
        float tw = logf(gb.z / aw + 1e-8f);
        float th = logf(gb.w / ah + 1e-8f);
        acc = smooth_l1(tv.x - tx) + smooth_l1(tv.y - ty)
            + smooth_l1(tv.z - tw) + smooth_l1(tv.w - th);
    }

    float s = block_reduce_256(acc, sh_red);
    if (threadIdx.x == 0) partial[blockIdx.x] = s;
}

/* Kernel 2: stream 47 MB of logits as float4, masked BCE, per-block partials */
__global__ __launch_bounds__(256)
void k_cls(const float4*   __restrict__ logits4,  /* NB*AA*NCLS/4 float4 */
           const uint16_t* __restrict__ meta,
           float*          __restrict__ partial)  /* [K2_BLOCKS] */
{
    const int N4 = NB * AA * NCLS / 4;            /* 2,949,120 — 80%4==0 so a float4 never crosses an anchor */
    const int stride = gridDim.x * 256;
    float acc = 0.0f;

    for (int i = blockIdx.x * 256 + threadIdx.x; i < N4; i += stride) {
        float4 v = logits4[i];
        int anchor = i / 20;                      /* 20 float4 per anchor row */
        int cbase  = (i - anchor * 20) * 4;
        unsigned m = meta[anchor];
        bool pos = (m & 256u) != 0u;
        bool neg = (m & 512u) != 0u;
        int  cat = (int)(m & 0xFFu);
        float xs[4] = { v.x, v.y, v.z, v.w };
        #pragma unroll
        for (int j = 0; j < 4; ++j) {
            float xv = xs[j];
            bool pen = (pos && (xv < HIGH_ENOUGH)) || (neg && (xv > LOW_ENOUGH));
            if (pen) {
                float t = (pos && (cbase + j) == cat) ? 1.0f : 0.0f;
                /* softplus tail via HW v_exp/v_log */
                float bce = fmaxf(xv, 0.0f) - xv * t
                          + __logf(1.0f + __expf(-fabsf(xv)));
                acc += bce;
            }
        }
    }

    __shared__ float sh[256];
    float s = block_reduce_256(acc, sh);
    if (threadIdx.x == 0) partial[blockIdx.x] = s;
}

/* Kernel 3: deterministic final sum, divide by nB */
__global__ __launch_bounds__(256)
void k_final(const float* __restrict__ p1, int n1,
             const float* __restrict__ p2, int n2,
             float* __restrict__ out)
{
    float acc = 0.0f;
    for (int i = threadIdx.x; i < n1; i += 256) acc += p1[i];
    for (int i = threadIdx.x; i < n2; i += 256) acc += p2[i];
    __shared__ float sh[256];
    float s = block_reduce_256(acc, sh);
    if (threadIdx.x == 0) out[0] = s * (1.0f / (float)NB);
}

extern "C" void kernel_launch(void* const* d_in, const int* in_sizes, int n_in,
                              void* d_out, int out_size, void* d_ws, size_t ws_size,
                              hipStream_t stream) {
    const float* t_xywh     = (const float*)d_in[0];   /* [4,9,64,64,4]  */
    const float* cls_logits = (const float*)d_in[1];   /* [4,9,64,64,80] */
    const float* gt_bboxes  = (const float*)d_in[2];   /* [4,32,4]       */
    const int*   gt_cats    = (const int*)  d_in[3];   /* [4,32]         */
    /* d_in[4]=img_h, d_in[5]=img_w unused (anchors depend only on nH,nW,STRIDE) */

    /* workspace layout: meta (u16 per anchor) | p1 | p2, all written before read */
    uint16_t* meta = (uint16_t*)d_ws;
    size_t meta_bytes = (size_t)NB * AA * sizeof(uint16_t);    /* 294912, 256-aligned */
    float* p1 = (float*)((char*)d_ws + meta_bytes);
    float* p2 = p1 + K1_BLOCKS;

    k_anchor<<<K1_BLOCKS, 256, 0, stream>>>((const float4*)t_xywh, gt_bboxes,
                                            gt_cats, meta, p1);
    k_cls   <<<K2_BLOCKS, 256, 0, stream>>>((const float4*)cls_logits, meta, p2);
    k_final <<<1, 256, 0, stream>>>(p1, K1_BLOCKS, p2, K2_BLOCKS, (float*)d_out);
}